// MambaMinimalBlock_50319836840443
// MI455X (gfx1250) — compile-verified
//
#include <hip/hip_runtime.h>
#include <hip/hip_bf16.h>
#include <math.h>

// MambaMinimalBlock forward for MI455X (gfx1250), f32 end-to-end.
// B=4, L=2048, D_MODEL=1024, D_INNER=2048, D_STATE=16, D_CONV=4. M = B*L = 8192.

typedef __attribute__((ext_vector_type(2))) float v2f;
typedef __attribute__((ext_vector_type(8))) float v8f;

#define BM 128
#define BN 128
#define KT 16
#define PADP 16              // extra pairs: kp-stride = (BM+PADP)*2 = 288 dwords ≡ 32 (mod 64)
#define MROW (BM + PADP)     // 144 pairs per kp row

// ---------------------------------------------------------------------------
// Tiled f32 WMMA GEMM:  Out[M,N] = A[M,K] @ W[N,K]^T   (+ optional softplus(+bias))
// 256 threads = 8 waves; block tile 128x128; wave tile 32(M) x 64(N) = 2x4
// v8f accumulators. LDS double-buffered, k-pair-interleaved layout so each
// WMMA fragment is one aligned ds_load_b64 straight into an even VGPR pair.
// Global loads are b128 and register-staged to overlap the next stage's
// traffic with the current stage's WMMAs; global_prefetch warms L2 two
// stages ahead. EPI: 0 = plain store, 1 = softplus(acc + bias[col]).
// ---------------------------------------------------------------------------
#define STORE_PAIRS(DST, buf, r, c, v)                          \
  {                                                             \
    DST[buf][((c) >> 1) + 0][r][0] = (v).x;                     \
    DST[buf][((c) >> 1) + 0][r][1] = (v).y;                     \
    DST[buf][((c) >> 1) + 1][r][0] = (v).z;                     \
    DST[buf][((c) >> 1) + 1][r][1] = (v).w;                     \
  }

template <int EPI>
__global__ __launch_bounds__(256) void gemm_wmma_f32(
    const float* __restrict__ A, const float* __restrict__ W,
    const float* __restrict__ bias, float* __restrict__ Out,
    int M, int N, int K) {
  // [buffer][k-pair][row][2]: pair kp holds k=2kp and k=2kp+1 contiguously.
  __shared__ float As[2][KT / 2][MROW][2];
  __shared__ float Bs[2][KT / 2][MROW][2];

  const int tid = threadIdx.x;
  const int lane = tid & 31;
  const int wave = tid >> 5;
  const int waveM = wave & 3;   // 4 waves along M: 4*32 = 128
  const int waveN = wave >> 2;  // 2 waves along N: 2*64 = 128
  const int m0 = blockIdx.y * BM;
  const int n0 = blockIdx.x * BN;
  const int lhalf = lane >> 4;  // 0: lanes 0-15 (K=kk,kk+1), 1: lanes 16-31 (K=kk+2,kk+3)
  const int l16 = lane & 15;

  // Cooperative-load coords: 4 lanes cover one row's 16 k's (64B coalesced).
  const int r0 = (tid + 0) >> 2, c0 = ((tid + 0) & 3) * 4;
  const int r1 = (tid + 256) >> 2, c1 = ((tid + 256) & 3) * 4;

  v8f acc[2][4] = {};
  float4 ra0, ra1, rb0, rb1;

  // Preload stage 0 into buffer 0.
  ra0 = *(const float4*)(A + (size_t)(m0 + r0) * K + c0);
  ra1 = *(const float4*)(A + (size_t)(m0 + r1) * K + c1);
  rb0 = *(const float4*)(W + (size_t)(n0 + r0) * K + c0);
  rb1 = *(const float4*)(W + (size_t)(n0 + r1) * K + c1);
  STORE_PAIRS(As, 0, r0, c0, ra0);
  STORE_PAIRS(As, 0, r1, c1, ra1);
  STORE_PAIRS(Bs, 0, r0, c0, rb0);
  STORE_PAIRS(Bs, 0, r1, c1, rb1);

  const int nstage = K / KT;
  for (int s = 0; s < nstage; ++s) {
    __syncthreads();  // stage s data visible in buffer (s&1)
    const int cur = s & 1;
    const bool more = (s + 1) < nstage;
    if (more) {
      const int kb = (s + 1) * KT;
      ra0 = *(const float4*)(A + (size_t)(m0 + r0) * K + kb + c0);
      ra1 = *(const float4*)(A + (size_t)(m0 + r1) * K + kb + c1);
      rb0 = *(const float4*)(W + (size_t)(n0 + r0) * K + kb + c0);
      rb1 = *(const float4*)(W + (size_t)(n0 + r1) * K + kb + c1);
      if (s + 2 < nstage) {
        __builtin_prefetch(A + (size_t)(m0 + r0) * K + (s + 2) * KT + c0, 0, 0);
        __builtin_prefetch(W + (size_t)(n0 + r0) * K + (s + 2) * KT + c0, 0, 0);
      }
    }

#pragma unroll
    for (int kk = 0; kk < KT; kk += 4) {
      const int pr = (kk >> 1) + lhalf;  // even pair for lanes 0-15, odd for 16-31
      v2f a[2], b[4];
#pragma unroll
      for (int i = 0; i < 2; ++i)
        a[i] = *(const v2f*)&As[cur][pr][waveM * 32 + i * 16 + l16][0];
#pragma unroll
      for (int j = 0; j < 4; ++j)
        b[j] = *(const v2f*)&Bs[cur][pr][waveN * 64 + j * 16 + l16][0];
#pragma unroll
      for (int i = 0; i < 2; ++i)
#pragma unroll
        for (int j = 0; j < 4; ++j)
          acc[i][j] = __builtin_amdgcn_wmma_f32_16x16x4_f32(
              /*neg_a=*/false, a[i], /*neg_b=*/false, b[j],
              /*c_mod=*/(short)0, acc[i][j], /*reuse_a=*/false, /*reuse_b=*/false);
    }

    if (more) {
      __syncthreads();  // everyone done reading buffer cur^1 (stage s-1)
      const int nxt = cur ^ 1;
      STORE_PAIRS(As, nxt, r0, c0, ra0);
      STORE_PAIRS(As, nxt, r1, c1, ra1);
      STORE_PAIRS(Bs, nxt, r0, c0, rb0);
      STORE_PAIRS(Bs, nxt, r1, c1, rb1);
    }
  }

  // C/D layout: lanes 0-15: N=l16, M=r; lanes 16-31: N=l16, M=8+r (vgpr r)
#pragma unroll
  for (int i = 0; i < 2; ++i) {
    const int mt = m0 + waveM * 32 + i * 16;
#pragma unroll
    for (int j = 0; j < 4; ++j) {
      const int col = n0 + waveN * 64 + j * 16 + l16;
      const float bv = (EPI == 1) ? bias[col] : 0.0f;
#pragma unroll
      for (int r = 0; r < 8; ++r) {
        const int row = mt + r + lhalf * 8;
        float v = acc[i][j][r];
        if (EPI == 1) {
          v += bv;
          v = (v > 20.0f) ? v : log1pf(__expf(v));  // softplus
        }
        Out[(size_t)row * N + col] = v;
      }
    }
  }
}

// ---------------------------------------------------------------------------
// Depthwise causal conv (d_conv=4, left pad 3) + SiLU.
// xz row stride 4096; xi is cols [0,2048). Output xc [8192 x 2048].
// ---------------------------------------------------------------------------
__global__ __launch_bounds__(256) void conv_silu_kernel(
    const float* __restrict__ xz, const float* __restrict__ conv_w,
    const float* __restrict__ conv_b, float* __restrict__ xc) {
  int idx = blockIdx.x * blockDim.x + threadIdx.x;  // over 8192*2048
  int e = idx & 2047;
  int m = idx >> 11;  // b*2048 + l
  int l = m & 2047;
  float s = conv_b[e];
#pragma unroll
  for (int j = 0; j < 4; ++j) {
    int ls = l - 3 + j;
    if (ls >= 0) s += conv_w[e * 4 + j] * xz[(size_t)(m - 3 + j) * 4096 + e];
  }
  float sig = 1.0f / (1.0f + __expf(-s));
  xc[(size_t)m * 2048 + e] = s * sig;
}

// ---------------------------------------------------------------------------
// x_proj: bc[m, s] = sum_k xc[m,k] * x_proj_w[s,k], s in [0,32). N=32 is too
// skinny for WMMA tiles to pay off; a dot kernel suffices (1.1 GFLOP).
// ---------------------------------------------------------------------------
__global__ __launch_bounds__(256) void bc_kernel(
    const float* __restrict__ xc, const float* __restrict__ xpw,
    float* __restrict__ bc) {
  int s = threadIdx.x & 31;
  int m = blockIdx.x * 8 + (threadIdx.x >> 5);
  const float* xr = xc + (size_t)m * 2048;
  const float* wr = xpw + (size_t)s * 2048;
  float a0 = 0.f, a1 = 0.f, a2 = 0.f, a3 = 0.f;
  for (int k = 0; k < 2048; k += 4) {
    a0 += xr[k + 0] * wr[k + 0];
    a1 += xr[k + 1] * wr[k + 1];
    a2 += xr[k + 2] * wr[k + 2];
    a3 += xr[k + 3] * wr[k + 3];
  }
  bc[(size_t)m * 32 + s] = (a0 + a1) + (a2 + a3);
}

// ---------------------------------------------------------------------------
// Selective scan. One half-wave (16 lanes) per channel (b,e); lane = state s.
// h_s recurrence in a register; y_t = cross-lane sum over 16 states
// (butterfly shfl_xor, width 16). Fused: + D*x_t, * silu(z).
// 131072 threads total -> 4096 waves of parallelism despite serial L.
// ---------------------------------------------------------------------------
__global__ __launch_bounds__(256) void scan_kernel(
    const float* __restrict__ delta, const float* __restrict__ xc,
    const float* __restrict__ bcb, const float* __restrict__ xz,
    const float* __restrict__ A_log, const float* __restrict__ Dp,
    float* __restrict__ y) {
  int t = blockIdx.x * blockDim.x + threadIdx.x;  // [0, 131072)
  int s = t & 15;
  int ch = t >> 4;  // b*2048 + e
  int e = ch & 2047;
  int b = ch >> 11;
  float a_s = -__expf(A_log[e * 16 + s]);  // A = -exp(A_log)
  float d_e = Dp[e];
  float h = 0.f;
  size_t mbase = (size_t)b * 2048;
  for (int l = 0; l < 2048; ++l) {
    size_t m = mbase + l;
    float dt = delta[m * 2048 + e];  // broadcast within half-wave
    float xt = xc[m * 2048 + e];
    float Bv = bcb[m * 32 + s];
    float Cv = bcb[m * 32 + 16 + s];
    h = __expf(dt * a_s) * h + (dt * Bv) * xt;
    float p = h * Cv;
    p += __shfl_xor(p, 1, 16);
    p += __shfl_xor(p, 2, 16);
    p += __shfl_xor(p, 4, 16);
    p += __shfl_xor(p, 8, 16);
    if (s == 0) {
      float yt = p + d_e * xt;
      float z = xz[m * 4096 + 2048 + e];
      float sig = 1.0f / (1.0f + __expf(-z));
      y[m * 2048 + e] = yt * (z * sig);
    }
  }
}

// ---------------------------------------------------------------------------
extern "C" void kernel_launch(void* const* d_in, const int* in_sizes, int n_in,
                              void* d_out, int out_size, void* d_ws, size_t ws_size,
                              hipStream_t stream) {
  const float* x          = (const float*)d_in[0];  // [4,2048,1024]
  const float* in_proj_w  = (const float*)d_in[1];  // [4096,1024]
  const float* conv_w     = (const float*)d_in[2];  // [2048,1,4]
  const float* conv_b     = (const float*)d_in[3];  // [2048]
  const float* x_proj_w   = (const float*)d_in[4];  // [32,2048]
  const float* dt_proj_w  = (const float*)d_in[5];  // [2048,2048]
  const float* dt_proj_b  = (const float*)d_in[6];  // [2048]
  const float* A_log      = (const float*)d_in[7];  // [2048,16]
  const float* Dp         = (const float*)d_in[8];  // [2048]
  const float* out_proj_w = (const float*)d_in[9];  // [1024,2048]
  float* out = (float*)d_out;                       // [4,2048,1024]

  const int Mr = 8192;  // B*L

  // Workspace layout (f32): xz[8192*4096] | xconv[8192*2048] | delta[8192*2048]
  //                         | bc[8192*32] | y[8192*2048]   (~337 MB)
  float* xz    = (float*)d_ws;
  float* xconv = xz + (size_t)Mr * 4096;
  float* delta = xconv + (size_t)Mr * 2048;
  float* bcb   = delta + (size_t)Mr * 2048;
  float* ybuf  = bcb + (size_t)Mr * 32;

  dim3 blk(256);

  // 1) xz = x @ in_proj_w^T   [8192 x 4096]
  gemm_wmma_f32<0><<<dim3(4096 / BN, Mr / BM), blk, 0, stream>>>(
      x, in_proj_w, nullptr, xz, Mr, 4096, 1024);

  // 2) xconv = silu(causal_dwconv(xi) + b)
  conv_silu_kernel<<<(Mr * 2048) / 256, blk, 0, stream>>>(xz, conv_w, conv_b, xconv);

  // 3) delta = softplus(xconv @ dt_proj_w^T + dt_proj_b)   [8192 x 2048]
  gemm_wmma_f32<1><<<dim3(2048 / BN, Mr / BM), blk, 0, stream>>>(
      xconv, dt_proj_w, dt_proj_b, delta, Mr, 2048, 2048);

  // 4) bc = xconv @ x_proj_w^T   [8192 x 32]
  bc_kernel<<<Mr / 8, blk, 0, stream>>>(xconv, x_proj_w, bcb);

  // 5) selective scan + skip + gate -> ybuf [8192 x 2048]
  scan_kernel<<<(Mr * 16) / 256, blk, 0, stream>>>(
      delta, xconv, bcb, xz, A_log, Dp, ybuf);

  // 6) out = ybuf @ out_proj_w^T   [8192 x 1024]
  gemm_wmma_f32<0><<<dim3(1024 / BN, Mr / BM), blk, 0, stream>>>(
      ybuf, out_proj_w, nullptr, out, Mr, 1024, 2048);
}